// PrunedKVAttention_7164005450347
// MI455X (gfx1250) — compile-verified
//
#include <hip/hip_runtime.h>

typedef float v2f __attribute__((ext_vector_type(2)));
typedef float v8f __attribute__((ext_vector_type(8)));

#define B_    4
#define T_    1024
#define C_    1024
#define H_    16
#define D_    64
#define KKEEP 391
#define KPAD  400
#define REC   64
#define TSEL  (T_ - REC)     // 960 candidate positions
#define KSEL  (KKEEP - REC)  // 327 selected among candidates

#define AS_STRIDE 20         // 8/16B-aligned pairs, conflict-free (m*20 mod 64 distinct)
#define BS_STRIDE 18         // 8B-aligned pairs, conflict-free (n*18 mod 64 distinct)
#define QS_STRIDE 66
#define SS_STRIDE 402

static __device__ __forceinline__ v8f wmma_f32(v2f a, v2f b, v8f c) {
  // D = A(16x4 f32) * B(4x16 f32) + C(16x16 f32)
  return __builtin_amdgcn_wmma_f32_16x16x4_f32(false, a, false, b, (short)0, c,
                                               false, false);
}

// ---------------------------------------------------------------------------
// fp32 WMMA GEMM: C[M,N] = A[M,K] @ B[K,N], row-major.
// Block tile 128x64, 256 threads (8 waves); wave w owns rows [w*16, w*16+16),
// 4 N-tiles of 16. K staged in chunks of 16. Bs kept n-major so every
// fragment is a single aligned ds_load_b64.
// ---------------------------------------------------------------------------
__global__ __launch_bounds__(256) void gemm_f32_wmma(
    const float* __restrict__ A, const float* __restrict__ Bm,
    float* __restrict__ Cm, int N, int K) {
  __shared__ float As[128 * AS_STRIDE];  // [m][k]
  __shared__ float Bs[64 * BS_STRIDE];   // [n][k]  (transposed)

  const int tid  = threadIdx.x;
  const int lane = tid & 31;
  const int wave = tid >> 5;            // 0..7
  const int hi   = (lane >> 4) & 1;     // 0: lanes 0-15, 1: lanes 16-31
  const int l15  = lane & 15;
  const int m0   = blockIdx.x * 128;
  const int n0   = blockIdx.y * 64;

  v8f acc[4];
  for (int t = 0; t < 4; ++t)
    for (int i = 0; i < 8; ++i) acc[t][i] = 0.0f;

  for (int kc = 0; kc < K; kc += 16) {
    // Stage A: 128x16, thread loads 8 consecutive floats (16B-aligned dst)
    {
      const int row = tid >> 1, k8 = (tid & 1) * 8;
      const float* src = A + (size_t)(m0 + row) * K + kc + k8;
      float* dst = &As[row * AS_STRIDE + k8];
#pragma unroll
      for (int j = 0; j < 8; ++j) dst[j] = src[j];
    }
    // Stage B: 16x64 from global (row-major), write transposed [n][k]
    {
      const int i4 = tid * 4, row = i4 >> 6, col = i4 & 63;  // row=k, col=n
      const float* src = Bm + (size_t)(kc + row) * N + n0 + col;
#pragma unroll
      for (int j = 0; j < 4; ++j) Bs[(col + j) * BS_STRIDE + row] = src[j];
    }
    __syncthreads();

#pragma unroll
    for (int koff = 0; koff < 16; koff += 4) {
      const int kb = koff + hi * 2;     // frag K base per ISA layout
      const v2f a = *(const v2f*)&As[(wave * 16 + l15) * AS_STRIDE + kb];
#pragma unroll
      for (int t = 0; t < 4; ++t) {
        const v2f b = *(const v2f*)&Bs[(t * 16 + l15) * BS_STRIDE + kb];
        acc[t] = wmma_f32(a, b, acc[t]);
      }
    }
    __syncthreads();
  }

  // Store D: VGPR r -> row r (lanes 0-15) / row r+8 (lanes 16-31), col = l15
#pragma unroll
  for (int t = 0; t < 4; ++t)
#pragma unroll
    for (int r = 0; r < 8; ++r) {
      const int row = m0 + wave * 16 + r + hi * 8;
      const int col = n0 + t * 16 + l15;
      Cm[(size_t)row * N + col] = acc[t][r];
    }
}

// ---------------------------------------------------------------------------
// last[bh, t] = dot(q[b,h,T-1,:], k[b,h,t,:])  (unscaled: monotonic for topk)
// ---------------------------------------------------------------------------
__global__ __launch_bounds__(256) void last_scores(
    const float* __restrict__ qkv, float* __restrict__ last) {
  __shared__ float ql[D_];
  const int bh = blockIdx.x, b = bh >> 4, h = bh & 15;
  const int tid = threadIdx.x;
  if (tid < D_)
    ql[tid] = qkv[(size_t)(b * T_ + (T_ - 1)) * (3 * C_) + h * D_ + tid];
  __syncthreads();
  for (int t = tid; t < T_; t += 256) {
    const float* kr = qkv + (size_t)(b * T_ + t) * (3 * C_) + C_ + h * D_;
    float s = 0.0f;
#pragma unroll 8
    for (int d = 0; d < D_; ++d) s += ql[d] * kr[d];
    last[bh * T_ + t] = s;
  }
}

// ---------------------------------------------------------------------------
// Select top-KSEL of last[bh, 0..TSEL) by exact rank (deterministic ties),
// append the REC recency indices. Output order irrelevant (softmax + weighted
// sum are permutation invariant over the kept set).
// ---------------------------------------------------------------------------
__global__ __launch_bounds__(256) void select_topk(
    const float* __restrict__ last, int* __restrict__ idx) {
  __shared__ float sc[TSEL];
  __shared__ int cnt;
  const int bh = blockIdx.x, tid = threadIdx.x;
  if (tid == 0) cnt = 0;
  for (int i = tid; i < TSEL; i += 256) sc[i] = last[bh * T_ + i];
  __syncthreads();
  for (int i = tid; i < TSEL; i += 256) {
    const float v = sc[i];
    int rank = 0;
    for (int j = 0; j < TSEL; ++j) {
      const float u = sc[j];
      rank += (u > v) || (u == v && j < i);
    }
    if (rank < KSEL) {
      const int slot = atomicAdd(&cnt, 1);
      idx[bh * KKEEP + slot] = i;
    }
  }
  __syncthreads();
  if (tid < REC) idx[bh * KKEEP + KSEL + tid] = TSEL + tid;
}

// ---------------------------------------------------------------------------
// Gather: Kbuf[bh][j][d] (row-major) and Vbuf[bh][d][j] (TRANSPOSED) so the
// attention P@V B-fragments {V[k][n], V[k+1][n]} are memory-consecutive.
// Pad j in [KKEEP, KPAD) with zeros.
// ---------------------------------------------------------------------------
__global__ __launch_bounds__(256) void gather_kv(
    const float* __restrict__ qkv, const int* __restrict__ idx,
    float* __restrict__ Kbuf, float* __restrict__ Vbuf) {
  const int bh = blockIdx.x, b = bh >> 4, h = bh & 15;
  // K: [j][d]
  for (int i = threadIdx.x; i < KPAD * D_; i += 256) {
    const int j = i >> 6, d = i & 63;
    float kv = 0.0f;
    if (j < KKEEP) {
      const int t = idx[bh * KKEEP + j];
      kv = qkv[(size_t)(b * T_ + t) * (3 * C_) + C_ + h * D_ + d];
    }
    Kbuf[(size_t)bh * KPAD * D_ + i] = kv;
  }
  // V transposed: [d][j]
  for (int i = threadIdx.x; i < D_ * KPAD; i += 256) {
    const int d = i / KPAD, j = i % KPAD;
    float vv = 0.0f;
    if (j < KKEEP) {
      const int t = idx[bh * KKEEP + j];
      vv = qkv[(size_t)(b * T_ + t) * (3 * C_) + 2 * C_ + h * D_ + d];
    }
    Vbuf[(size_t)bh * D_ * KPAD + i] = vv;
  }
}

// ---------------------------------------------------------------------------
// Pruned attention, one wave per (bh, 16-row q block).
// S = (q*scale) @ K_keep^T  -> softmax rows -> O = P @ V_keep
// ---------------------------------------------------------------------------
__global__ __launch_bounds__(32) void attn_kernel(
    const float* __restrict__ qkv, const float* __restrict__ Kbuf,
    const float* __restrict__ Vbuf, float* __restrict__ outh) {
  __shared__ float Q[16 * QS_STRIDE];
  __shared__ float S[16 * SS_STRIDE];

  const int lane = threadIdx.x;
  const int hi   = (lane >> 4) & 1;
  const int l15  = lane & 15;
  const int bh   = blockIdx.x >> 6;   // / (T/16)
  const int qblk = blockIdx.x & 63;
  const int b = bh >> 4, h = bh & 15;
  const float scale = 0.125f;         // 1/sqrt(64)

  // Stage scaled q tile [16 x 64]
  for (int i = lane; i < 16 * D_; i += 32) {
    const int m = i >> 6, d = i & 63;
    Q[m * QS_STRIDE + d] =
        qkv[(size_t)(b * T_ + qblk * 16 + m) * (3 * C_) + h * D_ + d] * scale;
  }
  __syncthreads();

  // q fragments for all 16 K-steps (K dim = 64)
  v2f qa[16];
#pragma unroll
  for (int kk = 0; kk < 16; ++kk)
    qa[kk] = *(const v2f*)&Q[l15 * QS_STRIDE + kk * 4 + hi * 2];

  const float* Kb = Kbuf + (size_t)bh * KPAD * D_;   // [j][d]
  const float* Vt = Vbuf + (size_t)bh * D_ * KPAD;   // [d][j]

  // Phase A: scores S[16, KPAD]
  for (int nt = 0; nt < KPAD / 16; ++nt) {
    v8f acc;
    for (int i = 0; i < 8; ++i) acc[i] = 0.0f;
    const int j = nt * 16 + l15;     // kept-index column owned by this lane
#pragma unroll
    for (int kk = 0; kk < 16; ++kk) {
      const v2f bb = *(const v2f*)&Kb[(size_t)j * D_ + kk * 4 + hi * 2];
      acc = wmma_f32(qa[kk], bb, acc);
    }
#pragma unroll
    for (int r = 0; r < 8; ++r)
      S[(r + hi * 8) * SS_STRIDE + nt * 16 + l15] = acc[r];
  }
  __syncthreads();

  // Phase B: per-row softmax over first KKEEP columns; zero the pad
  if (lane < 16) {
    float* row = &S[lane * SS_STRIDE];
    float mx = -3.402823e38f;
    for (int j = 0; j < KKEEP; ++j) mx = fmaxf(mx, row[j]);
    float sum = 0.0f;
    for (int j = 0; j < KKEEP; ++j) {
      const float e = __expf(row[j] - mx);
      row[j] = e;
      sum += e;
    }
    const float inv = 1.0f / sum;
    for (int j = 0; j < KKEEP; ++j) row[j] *= inv;
    for (int j = KKEEP; j < KPAD; ++j) row[j] = 0.0f;
  }
  __syncthreads();

  // Phase C: O[16,64] = P[16,KPAD] @ V[KPAD,64]  (V transposed in memory)
  v8f o[4];
  for (int t = 0; t < 4; ++t)
    for (int i = 0; i < 8; ++i) o[t][i] = 0.0f;
  for (int kk = 0; kk < KPAD / 4; ++kk) {
    const int kb = kk * 4 + hi * 2;
    const v2f a = *(const v2f*)&S[l15 * SS_STRIDE + kb];
#pragma unroll
    for (int nt = 0; nt < 4; ++nt) {
      // B[k=j][n=d] pair {Vt[d][kb], Vt[d][kb+1]} is contiguous
      const v2f bb = *(const v2f*)&Vt[(size_t)(nt * 16 + l15) * KPAD + kb];
      o[nt] = wmma_f32(a, bb, o[nt]);
    }
  }

  // Store into [B, T, C] head-merged layout
#pragma unroll
  for (int nt = 0; nt < 4; ++nt)
#pragma unroll
    for (int r = 0; r < 8; ++r) {
      const int t = qblk * 16 + r + hi * 8;
      const int c = h * D_ + nt * 16 + l15;
      outh[(size_t)(b * T_ + t) * C_ + c] = o[nt][r];
    }
}

// ---------------------------------------------------------------------------
extern "C" void kernel_launch(void* const* d_in, const int* in_sizes, int n_in,
                              void* d_out, int out_size, void* d_ws,
                              size_t ws_size, hipStream_t stream) {
  const float* x      = (const float*)d_in[0];
  const float* W_attn = (const float*)d_in[1];
  const float* W_proj = (const float*)d_in[2];
  float* out = (float*)d_out;

  char* ws = (char*)d_ws;
  size_t off = 0;
  float* qkv  = (float*)(ws + off); off += (size_t)B_ * T_ * 3 * C_ * 4;   // 48 MB
  float* last = (float*)(ws + off); off += (size_t)B_ * H_ * T_ * 4;       // 256 KB
  int*   idx  = (int*)(ws + off);   off += (size_t)B_ * H_ * KKEEP * 4;
  off = (off + 255) & ~(size_t)255;
  float* Kbuf = (float*)(ws + off); off += (size_t)B_ * H_ * KPAD * D_ * 4; // 6.4 MB
  float* Vbuf = (float*)(ws + off); off += (size_t)B_ * H_ * D_ * KPAD * 4; // 6.4 MB
  float* outh = (float*)(ws + off); off += (size_t)B_ * T_ * C_ * 4;        // 16 MB

  // 1) qkv = x @ W_attn   (M=4096, K=1024, N=3072)
  gemm_f32_wmma<<<dim3(4096 / 128, 3072 / 64), 256, 0, stream>>>(
      x, W_attn, qkv, 3 * C_, C_);
  // 2) last-row scores per (b,h)
  last_scores<<<B_ * H_, 256, 0, stream>>>(qkv, last);
  // 3) top-k index set
  select_topk<<<B_ * H_, 256, 0, stream>>>(last, idx);
  // 4) gather K_keep / V_keep (V transposed)
  gather_kv<<<B_ * H_, 256, 0, stream>>>(qkv, idx, Kbuf, Vbuf);
  // 5) pruned attention
  attn_kernel<<<B_ * H_ * (T_ / 16), 32, 0, stream>>>(qkv, Kbuf, Vbuf, outh);
  // 6) out = outh @ W_proj   (M=4096, K=1024, N=1024)
  gemm_f32_wmma<<<dim3(4096 / 128, 1024 / 64), 256, 0, stream>>>(
      outh, W_proj, out, C_, C_);
}